// GAT_14242111553928
// MI455X (gfx1250) — compile-verified
//
#include <hip/hip_runtime.h>
#include <hip/hip_bf16.h>
#include <math.h>

typedef __attribute__((ext_vector_type(2))) float v2f;
typedef __attribute__((ext_vector_type(8))) float v8f;

#define LEAK 0.2f

// ---- order-preserving float <-> uint encoding for atomicMax on floats ----
__device__ __forceinline__ unsigned fenc(float x) {
  unsigned b = __float_as_uint(x);
  return (b & 0x80000000u) ? ~b : (b | 0x80000000u);
}
__device__ __forceinline__ float fdec(unsigned u) {
  unsigned b = (u & 0x80000000u) ? (u & 0x7FFFFFFFu) : ~u;
  return __uint_as_float(b);
}
#define NEG_INF_ENC 0x007FFFFFu  // fenc(-inf)

// ---------------- fill kernels (graph-capture-safe init) ----------------
__global__ void fill_f32(float* __restrict__ p, float v, long long n) {
  long long i = blockIdx.x * (long long)blockDim.x + threadIdx.x;
  if (i < n) p[i] = v;
}
__global__ void fill_u32(unsigned* __restrict__ p, unsigned v, long long n) {
  long long i = blockIdx.x * (long long)blockDim.x + threadIdx.x;
  if (i < n) p[i] = v;
}

// ---------------- FP32 WMMA GEMM: C[M,NC] = A[M,K] @ B[K,NC] ----------------
// One wave32 per 16x16 output tile; K stepped by 4 with V_WMMA_F32_16X16X4_F32.
// A 16x4 layout: lane holds A[m][kh], A[m][kh+1], m = lane&15, kh = 2*(lane>>4).
// B 4x16 layout: lane holds B[kh][n], B[kh+1][n], n = lane&15.
// C/D: VGPR r, lane L -> row = 8*(L>>4)+r, col = L&15.
// Requires M % 16 == 0, K % 4 == 0, NC % 16 == 0 (true for this problem).
__global__ void wmma_gemm_f32(const float* __restrict__ A, const float* __restrict__ B,
                              float* __restrict__ C, int M, int K, int NC) {
  const int lane = threadIdx.x & 31;
  const int wave = (int)((blockIdx.x * (long long)blockDim.x + threadIdx.x) >> 5);
  const int tiles_n = NC >> 4;
  const int tiles_m = M >> 4;
  if (wave >= tiles_m * tiles_n) return;
  const int tm = wave / tiles_n;
  const int tn = wave - tm * tiles_n;
  const int row = (tm << 4) + (lane & 15);
  const int col = (tn << 4) + (lane & 15);
  const int kh  = (lane >> 4) << 1;

  v8f acc = {};
  const float* ap = A + (size_t)row * K + kh;
  const float* bp = B + (size_t)kh * NC + col;
  for (int k = 0; k < K; k += 4) {
    v2f a = *(const v2f*)(ap + k);              // A[m][k+kh], A[m][k+kh+1] contiguous
    v2f b;
    b.x = bp[(size_t)k * NC];                    // B[k+kh][n]
    b.y = bp[(size_t)(k + 1) * NC];              // B[k+kh+1][n]
    acc = __builtin_amdgcn_wmma_f32_16x16x4_f32(
        /*neg_a=*/false, a, /*neg_b=*/false, b,
        /*c_mod=*/(short)0, acc, /*reuse_a=*/false, /*reuse_b=*/false);
  }
  float* cp = C + (size_t)((tm << 4) + ((lane >> 4) << 3)) * NC + col;
#pragma unroll
  for (int r = 0; r < 8; ++r) cp[(size_t)r * NC] = acc[r];
}

// ---------------- per-node attention logits: dot(xh[n,h,:], att[h,:]) ----------------
__global__ void node_alpha(const float* __restrict__ xh, const float* __restrict__ as,
                           const float* __restrict__ ad, float* __restrict__ osrc,
                           float* __restrict__ odst, int N, int C) {
  int gid = blockIdx.x * blockDim.x + threadIdx.x;
  if (gid >= N * 8) return;
  int n = gid >> 3, h = gid & 7;
  const float* f = xh + (size_t)n * 8 * C + (size_t)h * C;
  float ss = 0.0f, sd = 0.0f;
  for (int c = 0; c < C; ++c) {
    float v = f[c];
    ss += v * as[h * C + c];
    sd += v * ad[h * C + c];
  }
  osrc[gid] = ss;
  odst[gid] = sd;
}

// ---------------- edge passes (items = E real edges + N self loops) ----------------
__device__ __forceinline__ void edge_sd(const int* __restrict__ srcs,
                                        const int* __restrict__ dsts,
                                        int item, int E, int& s, int& d) {
  if (item < E) { s = srcs[item]; d = dsts[item]; }
  else          { s = item - E;   d = s; }
}

__global__ void edge_max(const int* __restrict__ srcs, const int* __restrict__ dsts,
                         int E, int N,
                         const float* __restrict__ asrc, const float* __restrict__ adst,
                         unsigned* __restrict__ mx) {
  long long gid = blockIdx.x * (long long)blockDim.x + threadIdx.x;
  long long total = ((long long)E + N) * 8;
  if (gid >= total) return;
  int item = (int)(gid >> 3), h = (int)(gid & 7);
  int s, d; edge_sd(srcs, dsts, item, E, s, d);
  float e = asrc[s * 8 + h] + adst[d * 8 + h];
  e = e > 0.0f ? e : LEAK * e;
  atomicMax(mx + d * 8 + h, fenc(e));
}

__global__ void edge_sum(const int* __restrict__ srcs, const int* __restrict__ dsts,
                         int E, int N,
                         const float* __restrict__ asrc, const float* __restrict__ adst,
                         const unsigned* __restrict__ mx, float* __restrict__ denom) {
  long long gid = blockIdx.x * (long long)blockDim.x + threadIdx.x;
  long long total = ((long long)E + N) * 8;
  if (gid >= total) return;
  int item = (int)(gid >> 3), h = (int)(gid & 7);
  int s, d; edge_sd(srcs, dsts, item, E, s, d);
  float e = asrc[s * 8 + h] + adst[d * 8 + h];
  e = e > 0.0f ? e : LEAK * e;
  atomicAdd(denom + d * 8 + h, expf(e - fdec(mx[d * 8 + h])));
}

// layer-1 aggregation: thread per (edge, head), 8 channels each
__global__ void edge_agg_l1(const int* __restrict__ srcs, const int* __restrict__ dsts,
                            int E, int N,
                            const float* __restrict__ asrc, const float* __restrict__ adst,
                            const unsigned* __restrict__ mx, const float* __restrict__ denom,
                            const float* __restrict__ xh, float* __restrict__ agg) {
  long long gid = blockIdx.x * (long long)blockDim.x + threadIdx.x;
  long long total = ((long long)E + N) * 8;
  if (gid >= total) return;
  int item = (int)(gid >> 3), h = (int)(gid & 7);
  int s, d; edge_sd(srcs, dsts, item, E, s, d);
  float e = asrc[s * 8 + h] + adst[d * 8 + h];
  e = e > 0.0f ? e : LEAK * e;
  float alpha = expf(e - fdec(mx[d * 8 + h])) / (denom[d * 8 + h] + 1e-16f);
  const float* f = xh + (size_t)s * 64 + h * 8;
  float* o = agg + (size_t)d * 64 + h * 8;
#pragma unroll
  for (int c = 0; c < 8; ++c) atomicAdd(o + c, alpha * f[c]);
}

// layer-2 aggregation: block (128 threads) per edge.
// Stage xh2[src, 0:1024] (4 KB) into LDS with CDNA5 async global->LDS copies
// (ASYNCcnt path), overlap the 8 softmax-alpha exp computations with the copy,
// then fold heads in LDS and do one coalesced 128-wide atomic scatter.
__global__ void edge_agg_l2(const int* __restrict__ srcs, const int* __restrict__ dsts,
                            int E, int N,
                            const float* __restrict__ asrc, const float* __restrict__ adst,
                            const unsigned* __restrict__ mx, const float* __restrict__ denom,
                            const float* __restrict__ xh, float* __restrict__ out) {
  __shared__ __align__(16) float feat[1024];  // 4 KB: one source node's [8,128] features
  __shared__ float sal[8];
  int item = blockIdx.x;
  int s, d; edge_sd(srcs, dsts, item, E, s, d);
  int t = threadIdx.x;

  // --- async stage: 128 lanes x 16 B x 2 issues = 4096 B into LDS ---
  unsigned lds_addr = (unsigned)(unsigned long long)(void*)feat + (unsigned)(t * 16);
  unsigned long long gaddr =
      (unsigned long long)(xh + (size_t)s * 1024) + (unsigned long long)(t * 16);
  asm volatile(
      "global_load_async_to_lds_b128 %0, %1, off\n\t"
      "global_load_async_to_lds_b128 %0, %1, off offset:2048"
      :: "v"(lds_addr), "v"(gaddr)
      : "memory");

  // --- overlap: per-head softmax weights while the copy is in flight ---
  if (t < 8) {
    float e = asrc[s * 8 + t] + adst[d * 8 + t];
    e = e > 0.0f ? e : LEAK * e;
    sal[t] = expf(e - fdec(mx[d * 8 + t])) / (denom[d * 8 + t] + 1e-16f);
  }

  asm volatile("s_wait_asynccnt 0x0" ::: "memory");
  __syncthreads();

  float sum = 0.0f;
#pragma unroll
  for (int h = 0; h < 8; ++h) sum += sal[h] * feat[h * 128 + t];
  atomicAdd(out + (size_t)d * 128 + t, sum);
}

// ---------------- elementwise ----------------
__global__ void relu_bias64(float* __restrict__ h, const float* __restrict__ b,
                            long long n) {
  long long i = blockIdx.x * (long long)blockDim.x + threadIdx.x;
  if (i >= n) return;
  float v = h[i] + b[i & 63];
  h[i] = v > 0.0f ? v : 0.0f;
}
__global__ void finalize_mean_bias(float* __restrict__ out, const float* __restrict__ b,
                                   long long n) {
  long long i = blockIdx.x * (long long)blockDim.x + threadIdx.x;
  if (i >= n) return;
  out[i] = out[i] * 0.125f + b[i & 127];
}

// ---------------- launcher ----------------
static inline int cdiv_ll(long long a, long long b) { return (int)((a + b - 1) / b); }

extern "C" void kernel_launch(void* const* d_in, const int* in_sizes, int n_in,
                              void* d_out, int out_size, void* d_ws, size_t ws_size,
                              hipStream_t stream) {
  const float* x     = (const float*)d_in[0];
  const int*   ei    = (const int*)d_in[1];    // [2,E] row-major int32
  const float* W1    = (const float*)d_in[2];  // [128, 64]
  const float* attS1 = (const float*)d_in[3];  // [8, 8]
  const float* attD1 = (const float*)d_in[4];
  const float* b1    = (const float*)d_in[5];  // [64]
  const float* W2    = (const float*)d_in[6];  // [64, 1024]
  const float* attS2 = (const float*)d_in[7];  // [8, 128]
  const float* attD2 = (const float*)d_in[8];
  const float* b2    = (const float*)d_in[9];  // [128]

  const int N = in_sizes[0] / 128;
  const int E = in_sizes[1] / 2;
  const int* srcs = ei;
  const int* dsts = ei + E;

  // workspace carve-up (floats): xh1[N*64] asrc[N*8] adst[N*8] mx[N*8] denom[N*8]
  //                              agg1[N*64] xh2[N*1024]   ~= 237 MB total
  float*    xh1   = (float*)d_ws;
  float*    asrc  = xh1   + (size_t)N * 64;
  float*    adst  = asrc  + (size_t)N * 8;
  unsigned* mx    = (unsigned*)(adst + (size_t)N * 8);
  float*    denom = (float*)(mx + (size_t)N * 8);
  float*    agg1  = denom + (size_t)N * 8;
  float*    xh2   = agg1  + (size_t)N * 64;
  float*    out   = (float*)d_out;

  const int B = 256;
  const long long items = (long long)E + N;  // edges + self loops
  const long long itemsH = items * 8;

  // ================= layer 1 =================
  fill_f32<<<cdiv_ll((long long)N * 64, B), B, 0, stream>>>(agg1, 0.0f, (long long)N * 64);
  fill_f32<<<cdiv_ll((long long)N * 8, B), B, 0, stream>>>(denom, 0.0f, (long long)N * 8);
  fill_u32<<<cdiv_ll((long long)N * 8, B), B, 0, stream>>>(mx, NEG_INF_ENC, (long long)N * 8);

  {
    long long waves = (long long)(N / 16) * (64 / 16);
    wmma_gemm_f32<<<cdiv_ll(waves * 32, B), B, 0, stream>>>(x, W1, xh1, N, 128, 64);
  }
  node_alpha<<<cdiv_ll((long long)N * 8, B), B, 0, stream>>>(xh1, attS1, attD1, asrc, adst, N, 8);

  edge_max<<<cdiv_ll(itemsH, B), B, 0, stream>>>(srcs, dsts, E, N, asrc, adst, mx);
  edge_sum<<<cdiv_ll(itemsH, B), B, 0, stream>>>(srcs, dsts, E, N, asrc, adst, mx, denom);
  edge_agg_l1<<<cdiv_ll(itemsH, B), B, 0, stream>>>(srcs, dsts, E, N, asrc, adst, mx, denom,
                                                    xh1, agg1);
  relu_bias64<<<cdiv_ll((long long)N * 64, B), B, 0, stream>>>(agg1, b1, (long long)N * 64);

  // ================= layer 2 =================
  fill_f32<<<cdiv_ll((long long)N * 8, B), B, 0, stream>>>(denom, 0.0f, (long long)N * 8);
  fill_u32<<<cdiv_ll((long long)N * 8, B), B, 0, stream>>>(mx, NEG_INF_ENC, (long long)N * 8);
  fill_f32<<<cdiv_ll((long long)N * 128, B), B, 0, stream>>>(out, 0.0f, (long long)N * 128);

  {
    long long waves = (long long)(N / 16) * (1024 / 16);
    wmma_gemm_f32<<<cdiv_ll(waves * 32, B), B, 0, stream>>>(agg1, W2, xh2, N, 64, 1024);
  }
  node_alpha<<<cdiv_ll((long long)N * 8, B), B, 0, stream>>>(xh2, attS2, attD2, asrc, adst, N, 128);

  edge_max<<<cdiv_ll(itemsH, B), B, 0, stream>>>(srcs, dsts, E, N, asrc, adst, mx);
  edge_sum<<<cdiv_ll(itemsH, B), B, 0, stream>>>(srcs, dsts, E, N, asrc, adst, mx, denom);
  edge_agg_l2<<<(int)items, 128, 0, stream>>>(srcs, dsts, E, N, asrc, adst, mx, denom, xh2, out);

  finalize_mean_bias<<<cdiv_ll((long long)N * 128, B), B, 0, stream>>>(out, b2, (long long)N * 128);
}